// CenterNetDecoder_14439680049501
// MI455X (gfx1250) — compile-verified
//
#include <hip/hip_runtime.h>
#include <stdint.h>

// ---------------- problem constants (match reference) ----------------
#define B_      16
#define C_      8
#define H_      512
#define W_      512
#define HW_     (H_ * W_)       // 262144 = 2^18
#define K_      100
#define NDIR    12
#define PPM     4.0f
#define NBINS   4096            // 12-bit radix-select histogram
#define CANDMAX 2048            // max compacted candidates per batch
#define ROWS    16              // image rows per NMS tile (+1 halo row each side)
#define TPB     256

// ---------------- order-preserving float <-> uint --------------------
__device__ __forceinline__ unsigned f2ord(float f) {
  unsigned u = __float_as_uint(f);
  return (u & 0x80000000u) ? ~u : (u | 0x80000000u);
}
__device__ __forceinline__ float ord2f(unsigned u) {
  unsigned b = (u & 0x80000000u) ? (u & 0x7FFFFFFFu) : ~u;
  return __uint_as_float(b);
}

// ---------------- CDNA5 async global->LDS staging ---------------------
// clang prototype: (vNi AS(1)* gaddr, vNi AS(3)* ldsaddr, imm offset, imm cpol)
typedef int v4i_ __attribute__((vector_size(16)));
typedef __attribute__((address_space(1))) v4i_ g_v4i;
typedef __attribute__((address_space(3))) v4i_ l_v4i;
typedef int v2i_ __attribute__((vector_size(8)));
typedef __attribute__((address_space(1))) v2i_ g_v2i;
typedef __attribute__((address_space(3))) v2i_ l_v2i;

__device__ __forceinline__ void async_copy_b128(const float* g, float* l) {
#if __has_builtin(__builtin_amdgcn_global_load_async_to_lds_b128)
  __builtin_amdgcn_global_load_async_to_lds_b128(
      (g_v4i*)g, (l_v4i*)l, /*imm offset*/ 0, /*cpol*/ 0);
#elif __has_builtin(__builtin_amdgcn_global_load_async_to_lds_b64)
  __builtin_amdgcn_global_load_async_to_lds_b64((g_v2i*)g, (l_v2i*)l, 0, 0);
  __builtin_amdgcn_global_load_async_to_lds_b64((g_v2i*)(g + 2), (l_v2i*)(l + 2), 0, 0);
#else
  unsigned loff =
      (unsigned)(unsigned long long)(__attribute__((address_space(3))) void*)l;
  asm volatile("global_load_async_to_lds_b128 %0, %1, off"
               :: "v"(loff), "v"(g) : "memory");
#endif
}

__device__ __forceinline__ void wait_async0() {
#if __has_builtin(__builtin_amdgcn_s_wait_asynccnt)
  __builtin_amdgcn_s_wait_asynccnt(0);
#else
  asm volatile("s_wait_asynccnt 0" ::: "memory");
#endif
}

// Issue async copies for rows [y0-1, y0+ROWS] of one (b,c) plane into LDS:
// 18 rows x 512 floats = 9216 floats = 256 threads x 9 x (4 floats, b128).
// Halo rows outside the image are filled with -inf (SAME padding identity).
// Does NOT wait -- caller overlaps independent work, then calls tile_wait().
__device__ __forceinline__ void tile_issue(const float* __restrict__ plane,
                                           int y0, float* tile, int t) {
  const int xo = (t & 127) * 4;          // column offset, 16B aligned
  const int rh = t >> 7;                 // 0 or 1: which row of the pair
  #pragma unroll
  for (int i = 0; i < (ROWS + 2) / 2; ++i) {
    int r  = 2 * i + rh;                 // tile row 0..17
    int gy = y0 - 1 + r;
    float* dst = tile + r * W_ + xo;
    if (gy >= 0 && gy < H_) {
      async_copy_b128(plane + (size_t)gy * W_ + xo, dst);
    } else {
      dst[0] = dst[1] = dst[2] = dst[3] = -__builtin_inff();
    }
  }
}
__device__ __forceinline__ void tile_wait() {
  wait_async0();
  __syncthreads();
}

// Separable rolling 3x3 max scan: each thread owns columns {t, t+256} and
// walks the 16 pixel rows top-down, keeping per-row horizontal maxima in
// registers -> ~3 LDS loads/pixel instead of 9 (LDS bw is the limiter, not HBM).
// EMIT(v, gy, x) is called for every surviving local maximum.
#define NMS_SCAN(tile, t, y0, EMIT)                                            \
  do {                                                                         \
    _Pragma("unroll")                                                          \
    for (int half = 0; half < 2; ++half) {                                     \
      const int x = (t) + half * TPB;                                          \
      const bool has_l = (x > 0), has_r = (x < W_ - 1);                        \
      float hm_prev, hm_cur, hm_next, v_cur, v_next;                           \
      {                                                                        \
        const float* rp = (tile);                                              \
        float m = rp[x];                                                       \
        if (has_l) m = fmaxf(m, rp[x - 1]);                                    \
        if (has_r) m = fmaxf(m, rp[x + 1]);                                    \
        hm_prev = m;                                                           \
      }                                                                        \
      {                                                                        \
        const float* rp = (tile) + W_;                                         \
        v_cur = rp[x];                                                         \
        float m = v_cur;                                                       \
        if (has_l) m = fmaxf(m, rp[x - 1]);                                    \
        if (has_r) m = fmaxf(m, rp[x + 1]);                                    \
        hm_cur = m;                                                            \
      }                                                                        \
      _Pragma("unroll")                                                        \
      for (int ri = 1; ri <= ROWS; ++ri) {                                     \
        const float* rp = (tile) + (ri + 1) * W_;                              \
        v_next = rp[x];                                                        \
        float m = v_next;                                                      \
        if (has_l) m = fmaxf(m, rp[x - 1]);                                    \
        if (has_r) m = fmaxf(m, rp[x + 1]);                                    \
        hm_next = m;                                                           \
        float vm = fmaxf(fmaxf(hm_prev, hm_cur), hm_next);                     \
        if (v_cur >= vm) { EMIT(v_cur, ((y0) + ri - 1), x); }                  \
        hm_prev = hm_cur; hm_cur = hm_next; v_cur = v_next;                    \
      }                                                                        \
    }                                                                          \
  } while (0)

// ---------------- kernel 1: NMS + per-batch histogram ----------------
// (first in the TU so the disasm snippet shows the async-LDS path)
__global__ __launch_bounds__(TPB)
void k_hist(const float* __restrict__ heat, unsigned* __restrict__ hist) {
  __shared__ __align__(16) float tile[(ROWS + 2) * W_];   // 36 KB
  __shared__ unsigned lhist[NBINS];                       // 16 KB
  const int t  = threadIdx.x;
  const int c  = blockIdx.y;
  const int b  = blockIdx.z;
  const int y0 = blockIdx.x * ROWS;

  // kick off the tile DMA first; zero the histogram while copies are in flight
  const float* plane = heat + (size_t)(b * C_ + c) * HW_;
  tile_issue(plane, y0, tile, t);
  #pragma unroll
  for (int j = 0; j < NBINS / TPB; ++j) lhist[j * TPB + t] = 0u;
  tile_wait();                                 // asynccnt==0 + barrier

#define EMIT_HIST(v, gy, x) atomicAdd(&lhist[f2ord(v) >> 20], 1u)
  NMS_SCAN(tile, t, y0, EMIT_HIST);
#undef EMIT_HIST
  __syncthreads();

  unsigned* gh = hist + (size_t)b * NBINS;
  #pragma unroll
  for (int j = 0; j < NBINS / TPB; ++j) {
    unsigned n = lhist[j * TPB + t];
    if (n) atomicAdd(&gh[j * TPB + t], n);
  }
}

// ---------------- kernel 3: candidate compaction (hits 192MB L2) -----
__global__ __launch_bounds__(TPB)
void k_compact(const float* __restrict__ heat, const unsigned* __restrict__ thr,
               unsigned* __restrict__ cnt, unsigned long long* __restrict__ cand) {
  __shared__ __align__(16) float tile[(ROWS + 2) * W_];
  const int t  = threadIdx.x;
  const int c  = blockIdx.y;
  const int b  = blockIdx.z;
  const int y0 = blockIdx.x * ROWS;

  const float* plane = heat + (size_t)(b * C_ + c) * HW_;
  tile_issue(plane, y0, tile, t);
  const unsigned T = thr[b];                   // scalar load overlaps the DMA
  tile_wait();

#define EMIT_CAND(v, gy, x)                                                    \
  do {                                                                         \
    unsigned u_ = f2ord(v);                                                    \
    if ((u_ >> 20) >= T) {                                                     \
      unsigned pos_ = atomicAdd(&cnt[b], 1u);                                  \
      if (pos_ < CANDMAX) {                                                    \
        unsigned idx_ = ((unsigned)c << 18) | ((unsigned)(gy) << 9) |          \
                        (unsigned)(x);                                         \
        cand[(size_t)b * CANDMAX + pos_] =                                     \
            ((unsigned long long)u_ << 32) | (unsigned)(~idx_);                \
      }                                                                        \
    }                                                                          \
  } while (0)
  NMS_SCAN(tile, t, y0, EMIT_CAND);
#undef EMIT_CAND
}

// ---------------- kernel 2: per-batch threshold bin ------------------
__global__ void k_thresh(const unsigned* __restrict__ hist,
                         unsigned* __restrict__ thr) {
  int b = threadIdx.x;
  if (b >= B_) return;
  const unsigned* gh = hist + (size_t)b * NBINS;
  unsigned cum = 0, T = 0;
  for (int bin = NBINS - 1; bin >= 0; --bin) {
    cum += gh[bin];
    if (cum >= K_) { T = (unsigned)bin; break; }
  }
  thr[b] = T;
}

// ---------------- kernel 0: zero workspace ---------------------------
__global__ void k_init(unsigned* __restrict__ hist, unsigned* __restrict__ cnt) {
  int i = blockIdx.x * TPB + threadIdx.x;
  if (i < B_ * NBINS) hist[i] = 0u;
  if (i < B_)         cnt[i]  = 0u;
}

// ---------------- kernel 4: exact top-100 + decode + gather ----------
__global__ __launch_bounds__(TPB)
void k_select(const unsigned* __restrict__ cnt,
              const unsigned long long* __restrict__ cand,
              const float* __restrict__ wh, const float* __restrict__ off,
              const float* __restrict__ yawc, const float* __restrict__ yawr,
              const float* __restrict__ vel, float* __restrict__ out) {
  __shared__ unsigned long long wmax[TPB / 32];
  __shared__ unsigned long long winner_s;
  __shared__ unsigned long long topk[K_];
  const int t = threadIdx.x;
  const int b = blockIdx.x;

  unsigned n = cnt[b];
  if (n > CANDMAX) n = CANDMAX;
  const unsigned long long* cb = cand + (size_t)b * CANDMAX;

  unsigned long long regs[CANDMAX / TPB];
  #pragma unroll
  for (int j = 0; j < CANDMAX / TPB; ++j) {
    int i = j * TPB + t;
    regs[j] = (i < (int)n) ? cb[i] : 0ull;
  }

  for (int it = 0; it < K_; ++it) {
    unsigned long long lm = 0ull;
    #pragma unroll
    for (int j = 0; j < CANDMAX / TPB; ++j) lm = regs[j] > lm ? regs[j] : lm;
    #pragma unroll
    for (int m = 16; m >= 1; m >>= 1) {           // wave32 reduction
      unsigned long long o = __shfl_xor(lm, m, 32);
      lm = o > lm ? o : lm;
    }
    if ((t & 31) == 0) wmax[t >> 5] = lm;
    __syncthreads();
    if (t == 0) {
      unsigned long long w = 0ull;
      #pragma unroll
      for (int i = 0; i < TPB / 32; ++i) w = wmax[i] > w ? wmax[i] : w;
      winner_s = w;
      topk[it] = w;
    }
    __syncthreads();
    unsigned long long w = winner_s;
    if (w) {
      #pragma unroll
      for (int j = 0; j < CANDMAX / TPB; ++j)
        if (regs[j] == w) regs[j] = 0ull;         // keys unique (idx in key)
    }
  }
  __syncthreads();

  if (t < K_) {
    unsigned long long key = topk[t];
    float* o = out + ((size_t)b * K_ + t) * 9;
    if (key == 0ull) {
      #pragma unroll
      for (int j = 0; j < 9; ++j) o[j] = 0.f;
    } else {
      unsigned u   = (unsigned)(key >> 32);
      unsigned idx = ~(unsigned)(key & 0xFFFFFFFFull);
      float logit  = ord2f(u);
      float score  = 1.f / (1.f + expf(-logit));  // sigmoid deferred to here
      int c = (int)(idx >> 18);
      int s = (int)(idx & (HW_ - 1));
      int y = s >> 9;
      int x = s & (W_ - 1);

      const float* whb = wh  + (size_t)b * 2 * HW_;
      const float* ofb = off + (size_t)b * 2 * HW_;
      float w0 = whb[s], w1 = whb[HW_ + s];
      float o0 = ofb[s], o1 = ofb[HW_ + s];

      const float* ycb = yawc + (size_t)b * NDIR * HW_;
      int   cls  = 0;
      float best = ycb[s];
      #pragma unroll
      for (int j = 1; j < NDIR; ++j) {
        float vj = ycb[(size_t)j * HW_ + s];
        if (vj > best) { best = vj; cls = j; }    // argmax, ties -> lowest j
      }
      float yr = yawr[(size_t)b * HW_ + s];
      float vl = vel [(size_t)b * HW_ + s];

      const float PI = 3.14159265358979323846f;
      float yaw = (float)cls * (2.f * PI / (float)NDIR) + yr;
      if (yaw > PI) yaw -= 2.f * PI;

      float xs = (float)x + o0;
      float ys = (float)y + o1;
      o[0] = xs * PPM; o[1] = ys * PPM; o[2] = w0 * PPM; o[3] = w1 * PPM;
      o[4] = yaw;      o[5] = vl;       o[6] = 0.f;
      o[7] = (float)c; o[8] = score;
    }
  }
}

// ---------------- host launcher --------------------------------------
extern "C" void kernel_launch(void* const* d_in, const int* in_sizes, int n_in,
                              void* d_out, int out_size, void* d_ws, size_t ws_size,
                              hipStream_t stream) {
  (void)in_sizes; (void)n_in; (void)out_size; (void)ws_size;
  const float* heat = (const float*)d_in[0];
  const float* wh   = (const float*)d_in[1];
  const float* off  = (const float*)d_in[2];
  const float* yawc = (const float*)d_in[3];
  const float* yawr = (const float*)d_in[4];
  const float* vel  = (const float*)d_in[5];
  // d_in[6] is k == 100, baked in as K_.
  float* out = (float*)d_out;

  char* ws = (char*)d_ws;                         // ~513 KB used
  unsigned* hist = (unsigned*)ws;                                   // B*NBINS u32
  unsigned* cnt  = (unsigned*)(ws + (size_t)B_ * NBINS * 4);        // B u32
  unsigned* thr  = (unsigned*)(ws + (size_t)B_ * NBINS * 4 + 256);  // B u32
  unsigned long long* cand =
      (unsigned long long*)(ws + (size_t)B_ * NBINS * 4 + 512);     // B*CANDMAX u64

  k_init<<<(B_ * NBINS + TPB - 1) / TPB, TPB, 0, stream>>>(hist, cnt);

  dim3 grid(H_ / ROWS, C_, B_);                   // 32 x 8 x 16 = 4096 blocks
  k_hist<<<grid, TPB, 0, stream>>>(heat, hist);
  k_thresh<<<1, 32, 0, stream>>>(hist, thr);
  k_compact<<<grid, TPB, 0, stream>>>(heat, thr, cnt, cand);
  k_select<<<B_, TPB, 0, stream>>>(cnt, cand, wh, off, yawc, yawr, vel, out);
}